// TrigonometryUpdate_22574348108022
// MI455X (gfx1250) — compile-verified
//
#include <hip/hip_runtime.h>
#include <math.h>

typedef float v2f __attribute__((ext_vector_type(2)));
typedef float v4f __attribute__((ext_vector_type(4)));
typedef float v8f __attribute__((ext_vector_type(8)));
typedef int   v4i_g __attribute__((vector_size(16)));   // matches builtin param

#define AS1 __attribute__((address_space(1)))
#define AS3 __attribute__((address_space(3)))

#define ECH 256          // embedding channels
#define CCH 128          // c channels
#define NN  512          // n1 == n2
#define MTOT (512*512)   // z rows
#define NEDGE 16384
#define LN_EPS 1e-5f

#if defined(__has_builtin)
#if __has_builtin(__builtin_amdgcn_global_load_async_to_lds_b128) && \
    __has_builtin(__builtin_amdgcn_global_load_async_to_lds_b32)
#define HAS_ASYNC_LDS 1
#endif
#endif
#ifndef HAS_ASYNC_LDS
#define HAS_ASYNC_LDS 0
#endif

__device__ __forceinline__ float sigmoidf_(float x) {
    return 1.0f / (1.0f + __expf(-x));
}

// D = A(16x4, f32) x B(4x16, f32) + C   -- full fp32 WMMA
__device__ __forceinline__ v8f wmma_f32(v2f a, v2f b, v8f c) {
    return __builtin_amdgcn_wmma_f32_16x16x4_f32(
        /*neg_a=*/false, a, /*neg_b=*/false, b,
        /*c_mod=*/(short)0, c, /*reuse_a=*/false, /*reuse_b=*/false);
}

__device__ __forceinline__ v8f zero8() {
    v8f z = {0.f,0.f,0.f,0.f,0.f,0.f,0.f,0.f};
    return z;
}

// Async global->LDS copies (CDNA5 ASYNCcnt path) with synchronous fallback.
__device__ __forceinline__ void cp_lds_b128(const float* g, float* l) {
#if HAS_ASYNC_LDS
    __builtin_amdgcn_global_load_async_to_lds_b128(
        (AS1 v4i_g*)(g), (AS3 v4i_g*)(l), 0, 0);
#else
    *(v4f*)l = *(const v4f*)g;
#endif
}
__device__ __forceinline__ void cp_lds_b32(const float* g, float* l) {
#if HAS_ASYNC_LDS
    __builtin_amdgcn_global_load_async_to_lds_b32(
        (AS1 int*)(g), (AS3 int*)(l), 0, 0);
#else
    *l = *g;
#endif
}
__device__ __forceinline__ void async_drain() {
#if HAS_ASYNC_LDS
    asm volatile("s_wait_asynccnt 0x0" ::: "memory");
#endif
}

// ---------------------------------------------------------------------------
// LayerNorm over last dim (E=256). One wave (32 lanes) per row, 8 rows/block.
// ---------------------------------------------------------------------------
__global__ void ln_rows_kernel(const float* __restrict__ X,
                               const float* __restrict__ w,
                               const float* __restrict__ b,
                               float* __restrict__ Y, int rows) {
    int wave = threadIdx.x >> 5;
    int lane = threadIdx.x & 31;
    int row  = blockIdx.x * 8 + wave;
    if (row >= rows) return;
    const float* xr = X + (size_t)row * ECH;
    float vals[8];
    float s = 0.0f;
#pragma unroll
    for (int j = 0; j < 8; ++j) { float v = xr[lane + 32 * j]; vals[j] = v; s += v; }
#pragma unroll
    for (int off = 16; off > 0; off >>= 1) s += __shfl_xor(s, off);
    float mean = s * (1.0f / ECH);
    float sq = 0.0f;
#pragma unroll
    for (int j = 0; j < 8; ++j) { float d = vals[j] - mean; sq += d * d; }
#pragma unroll
    for (int off = 16; off > 0; off >>= 1) sq += __shfl_xor(sq, off);
    float inv = rsqrtf(sq * (1.0f / ECH) + LN_EPS);
    float* yr = Y + (size_t)row * ECH;
#pragma unroll
    for (int j = 0; j < 8; ++j) {
        int col = lane + 32 * j;
        yr[col] = (vals[j] - mean) * inv * w[col] + b[col];
    }
}

// ---------------------------------------------------------------------------
// Projection via WMMA.
//   GATED=true : out = sigmoid(X Wg^T + bg) * (X Wl^T + bl)
//   GATED=false: out = sigmoid(X Wg^T + bg)
// X: [M][256] row-major.  Wg/Wl: [N][256] (N-major, K=256).
// chan_major=1 -> OUT[n*M + m], else OUT[m*N + n].
// Wave computes a 32x16 tile; register double-buffering across K.
// ---------------------------------------------------------------------------
template<bool GATED>
__global__ void proj_gemm_kernel(const float* __restrict__ X, int M,
                                 const float* __restrict__ Wg, const float* __restrict__ bg,
                                 const float* __restrict__ Wl, const float* __restrict__ bl,
                                 float* __restrict__ OUT, int N, int chan_major) {
    const int K = ECH;
    int wave = threadIdx.x >> 5, lane = threadIdx.x & 31;
    int half = lane >> 4, idx = lane & 15;
    int m0 = (blockIdx.x * 8 + wave) * 32;
    int n0 = blockIdx.y * 16;

    const float* ar0 = X + (size_t)(m0 + idx) * K + 2 * half;
    const float* ar1 = X + (size_t)(m0 + 16 + idx) * K + 2 * half;
    const float* wgp = Wg + (size_t)(n0 + idx) * K + 2 * half;
    const float* wlp = GATED ? (Wl + (size_t)(n0 + idx) * K + 2 * half) : nullptr;

    v8f accg0 = zero8(), accg1 = zero8();
    v8f accl0 = zero8(), accl1 = zero8();

    v2f a0c = *(const v2f*)(ar0);
    v2f a1c = *(const v2f*)(ar1);
    v2f gc  = *(const v2f*)(wgp);
    v2f lc  = GATED ? *(const v2f*)(wlp) : a0c;

    for (int k = 0; k < K; k += 4) {
        int kn = (k + 4 < K) ? (k + 4) : 0;
        v2f a0n = *(const v2f*)(ar0 + kn);
        v2f a1n = *(const v2f*)(ar1 + kn);
        v2f gn  = *(const v2f*)(wgp + kn);
        v2f ln_ = GATED ? *(const v2f*)(wlp + kn) : a0n;

        accg0 = wmma_f32(a0c, gc, accg0);
        accg1 = wmma_f32(a1c, gc, accg1);
        if (GATED) {
            accl0 = wmma_f32(a0c, lc, accl0);
            accl1 = wmma_f32(a1c, lc, accl1);
        }
        a0c = a0n; a1c = a1n; gc = gn; lc = ln_;
    }

    int col = n0 + idx;
#pragma unroll
    for (int v = 0; v < 8; ++v) {
        int r = v + 8 * half;
        float g0 = sigmoidf_(accg0[v] + bg[col]);
        float g1 = sigmoidf_(accg1[v] + bg[col]);
        float o0 = GATED ? g0 * (accl0[v] + bl[col]) : g0;
        float o1 = GATED ? g1 * (accl1[v] + bl[col]) : g1;
        int row0 = m0 + r, row1 = m0 + 16 + r;
        if (chan_major) {
            OUT[(size_t)col * M + row0] = o0;
            OUT[(size_t)col * M + row1] = o1;
        } else {
            OUT[(size_t)row0 * N + col] = o0;
            OUT[(size_t)row1 * N + col] = o1;
        }
    }
}

// ---------------------------------------------------------------------------
// Scatter-add edge features into dense channel-major P[c][i][k].
// ---------------------------------------------------------------------------
__global__ void scatter_kernel(const int* __restrict__ edges,   // [2][ne]
                               const float* __restrict__ p,     // [ne][128]
                               float* __restrict__ P, int ne) {
    int e = blockIdx.x;
    int c = threadIdx.x;
    int i = edges[e];
    int k = edges[ne + e];
    atomicAdd(&P[(size_t)c * MTOT + (size_t)i * NN + k], p[(size_t)e * CCH + c]);
}

// ---------------------------------------------------------------------------
// Per-channel 512x512x512 fp32 GEMM:  C[c] (+)= A[c] @ B[c](^T).
// LDS-staged, double-buffered: block tile 128x128, K-step 16. Stage K-step
// t+1 via async global->LDS copies while WMMAs consume K-step t. B tile is
// always staged as [n][k] (the non-transposed case transposes during the
// per-lane async copy), so fragment reads are uniform ds b64 loads.
// Wave tile 64x32 (4x2 WMMA tiles): 8 WMMAs per 6 LDS fragment loads.
// LDS row stride 20 floats: b128 stores 16B-aligned, b64 reads 8B-aligned,
// 20*i mod 64 spans 16 banks (conflict-light).
// ---------------------------------------------------------------------------
#define KT 16
#define LSTR 20

template<bool B_TRANS, bool ACC>
__global__ void chan_gemm_kernel(const float* __restrict__ A,
                                 const float* __restrict__ Bm,
                                 float* __restrict__ C) {
    const int LD = NN;
    __shared__ float At[2][128][LSTR];
    __shared__ float Bt[2][128][LSTR];

    size_t base = (size_t)blockIdx.z * MTOT;
    int tid  = threadIdx.x;
    int wave = tid >> 5, lane = tid & 31;
    int half = lane >> 4, idx = lane & 15;
    int m0b = blockIdx.x * 128;
    int n0b = blockIdx.y * 128;
    int mw  = (wave >> 2) * 64;   // wave row offset within block tile
    int nw  = (wave & 3) * 32;    // wave col offset within block tile

    const float* Ab = A  + base;
    const float* Bb = Bm + base;

    // ---- staging helpers (cooperative across 256 threads) ----
    auto stage = [&](int buf, int kb) {
        // A tile: 128 rows x 16 k, 4-float chunks, 512 chunks total.
#pragma unroll
        for (int q = 0; q < 2; ++q) {
            int chunk = tid + q * 256;
            int r = chunk >> 2, c4 = (chunk & 3) * 4;
            cp_lds_b128(Ab + (size_t)(m0b + r) * LD + kb + c4, &At[buf][r][c4]);
        }
        if (B_TRANS) {
            // B stored [N][K]: same pattern as A.
#pragma unroll
            for (int q = 0; q < 2; ++q) {
                int chunk = tid + q * 256;
                int r = chunk >> 2, c4 = (chunk & 3) * 4;
                cp_lds_b128(Bb + (size_t)(n0b + r) * LD + kb + c4, &Bt[buf][r][c4]);
            }
        } else {
            // B stored [K][N]: transpose during staging -> LDS [n][k].
#pragma unroll
            for (int q = 0; q < 8; ++q) {
                int e = tid + q * 256;          // 0..2047
                int kk = e >> 7, n = e & 127;
                cp_lds_b32(Bb + (size_t)(kb + kk) * LD + n0b + n, &Bt[buf][n][kk]);
            }
        }
    };

    v8f acc[4][2];
#pragma unroll
    for (int i = 0; i < 4; ++i) { acc[i][0] = zero8(); acc[i][1] = zero8(); }

    stage(0, 0);
    async_drain();
    __syncthreads();

    int buf = 0;
    for (int kb = 0; kb < LD; kb += KT) {
        if (kb + KT < LD) stage(buf ^ 1, kb + KT);

#pragma unroll
        for (int k4 = 0; k4 < KT; k4 += 4) {
            int kk = k4 + 2 * half;
            v2f af[4], bf[2];
#pragma unroll
            for (int i = 0; i < 4; ++i)
                af[i] = *(const v2f*)&At[buf][mw + 16 * i + idx][kk];
#pragma unroll
            for (int j = 0; j < 2; ++j)
                bf[j] = *(const v2f*)&Bt[buf][nw + 16 * j + idx][kk];
#pragma unroll
            for (int i = 0; i < 4; ++i) {
                acc[i][0] = wmma_f32(af[i], bf[0], acc[i][0]);
                acc[i][1] = wmma_f32(af[i], bf[1], acc[i][1]);
            }
        }

        async_drain();
        __syncthreads();
        buf ^= 1;
    }

    float* Cb = C + base;
#pragma unroll
    for (int i = 0; i < 4; ++i) {
#pragma unroll
        for (int v = 0; v < 8; ++v) {
            int row = m0b + mw + 16 * i + v + 8 * half;
            size_t o0 = (size_t)row * LD + n0b + nw + idx;
            size_t o1 = o0 + 16;
            if (ACC) { Cb[o0] += acc[i][0][v]; Cb[o1] += acc[i][1][v]; }
            else     { Cb[o0]  = acc[i][0][v]; Cb[o1]  = acc[i][1][v]; }
        }
    }
}

// ---------------------------------------------------------------------------
// Final: LN over C=128 of the block sum (channel-major), then WMMA GEMM with
// las_w [256][128], fused with g-gating:  out = g * (ln(B) @ las_w^T + las_b).
// ---------------------------------------------------------------------------
__global__ void final_kernel(const float* __restrict__ B,      // [128][262144]
                             const float* __restrict__ g,      // [262144][256]
                             const float* __restrict__ lnc_w,
                             const float* __restrict__ lnc_b,
                             const float* __restrict__ las_w,  // [256][128]
                             const float* __restrict__ las_b,
                             float* __restrict__ out) {        // [262144][256]
    __shared__ float s[16][CCH + 8];
    int m0 = blockIdx.x * 16;
    int tid = threadIdx.x;

    for (int e = tid; e < 16 * CCH; e += 256) {
        int c = e >> 4, r = e & 15;
        s[r][c] = B[(size_t)c * MTOT + m0 + r];
    }
    __syncthreads();

    // LayerNorm over channels: 16 lanes per row (xor offsets stay in-half).
    int row = tid >> 4, t = tid & 15;
    float sum = 0.0f;
#pragma unroll
    for (int j = 0; j < 8; ++j) sum += s[row][t + 16 * j];
#pragma unroll
    for (int off = 8; off > 0; off >>= 1) sum += __shfl_xor(sum, off);
    float mean = sum * (1.0f / CCH);
    float sq = 0.0f;
#pragma unroll
    for (int j = 0; j < 8; ++j) { float d = s[row][t + 16 * j] - mean; sq += d * d; }
#pragma unroll
    for (int off = 8; off > 0; off >>= 1) sq += __shfl_xor(sq, off);
    float inv = rsqrtf(sq * (1.0f / CCH) + LN_EPS);
#pragma unroll
    for (int j = 0; j < 8; ++j) {
        int c = t + 16 * j;
        s[row][c] = (s[row][c] - mean) * inv * lnc_w[c] + lnc_b[c];
    }
    __syncthreads();

    int wave = tid >> 5, lane = tid & 31, half = lane >> 4, idx = lane & 15;
#pragma unroll
    for (int tt = 0; tt < 2; ++tt) {
        int n0 = (wave + tt * 8) * 16;
        const float* wp = las_w + (size_t)(n0 + idx) * CCH + 2 * half;
        v8f acc = zero8();
        v2f bc = *(const v2f*)(wp);
        for (int k = 0; k < CCH; k += 4) {
            int kn = (k + 4 < CCH) ? (k + 4) : 0;
            v2f bnx = *(const v2f*)(wp + kn);
            v2f af; af.x = s[idx][k + 2 * half]; af.y = s[idx][k + 2 * half + 1];
            acc = wmma_f32(af, bc, acc);
            bc = bnx;
        }
        int col = n0 + idx;
#pragma unroll
        for (int v = 0; v < 8; ++v) {
            int m = m0 + v + 8 * half;
            size_t oi = (size_t)m * ECH + col;
            out[oi] = g[oi] * (acc[v] + las_b[col]);
        }
    }
}

// ---------------------------------------------------------------------------
extern "C" void kernel_launch(void* const* d_in, const int* in_sizes, int n_in,
                              void* d_out, int out_size, void* d_ws, size_t ws_size,
                              hipStream_t stream) {
    (void)in_sizes; (void)n_in; (void)out_size; (void)ws_size;

    const float* z      = (const float*)d_in[0];
    const float* efeat1 = (const float*)d_in[1];
    const float* efeat2 = (const float*)d_in[2];
    const float* ln_w   = (const float*)d_in[3];
    const float* ln_b   = (const float*)d_in[4];
    const float* lnc_w  = (const float*)d_in[5];
    const float* lnc_b  = (const float*)d_in[6];
    const float* gl1_w  = (const float*)d_in[7];
    const float* gl1_b  = (const float*)d_in[8];
    const float* gl2_w  = (const float*)d_in[9];
    const float* gl2_b  = (const float*)d_in[10];
    const float* l1_w   = (const float*)d_in[11];
    const float* l1_b   = (const float*)d_in[12];
    const float* l2_w   = (const float*)d_in[13];
    const float* l2_b   = (const float*)d_in[14];
    const float* eg_w   = (const float*)d_in[15];
    const float* eg_b   = (const float*)d_in[16];
    const float* las_w  = (const float*)d_in[17];
    const float* las_b  = (const float*)d_in[18];
    const int*   edges1 = (const int*)d_in[19];
    const int*   edges2 = (const int*)d_in[20];
    float* out = (float*)d_out;

    float* ws = (float*)d_ws;
    size_t off = 0;
    float* zl   = ws + off; off += (size_t)MTOT * ECH;   // LN(z)
    float* gbuf = ws + off; off += (size_t)MTOT * ECH;   // sigmoid gate g
    float* ab1  = ws + off; off += (size_t)CCH * MTOT;   // channel-major
    float* ab2  = ws + off; off += (size_t)CCH * MTOT;
    float* P1   = ws + off; off += (size_t)CCH * MTOT;
    float* P2   = ws + off; off += (size_t)CCH * MTOT;
    float* Bsum = ws + off; off += (size_t)CCH * MTOT;   // block1 + block2
    float* pl1  = ws + off; off += (size_t)NEDGE * ECH;  // LN(efeat1)
    float* pl2  = ws + off; off += (size_t)NEDGE * ECH;
    float* p1   = ws + off; off += (size_t)NEDGE * CCH;  // gated edge feats
    float* p2   = ws + off; off += (size_t)NEDGE * CCH;

    // 1) LayerNorms over E
    ln_rows_kernel<<<MTOT / 8, 256, 0, stream>>>(z, ln_w, ln_b, zl, MTOT);
    ln_rows_kernel<<<NEDGE / 8, 256, 0, stream>>>(efeat1, ln_w, ln_b, pl1, NEDGE);
    ln_rows_kernel<<<NEDGE / 8, 256, 0, stream>>>(efeat2, ln_w, ln_b, pl2, NEDGE);

    // 2) Gated projections (WMMA), channel-major for z; edge gate g
    proj_gemm_kernel<true><<<dim3(MTOT / 256, CCH / 16), 256, 0, stream>>>(
        zl, MTOT, gl1_w, gl1_b, l1_w, l1_b, ab1, CCH, /*chan_major=*/1);
    proj_gemm_kernel<true><<<dim3(MTOT / 256, CCH / 16), 256, 0, stream>>>(
        zl, MTOT, gl2_w, gl2_b, l2_w, l2_b, ab2, CCH, /*chan_major=*/1);
    proj_gemm_kernel<false><<<dim3(MTOT / 256, ECH / 16), 256, 0, stream>>>(
        zl, MTOT, eg_w, eg_b, nullptr, nullptr, gbuf, ECH, /*chan_major=*/0);
    proj_gemm_kernel<true><<<dim3(NEDGE / 256, CCH / 16), 256, 0, stream>>>(
        pl1, NEDGE, gl1_w, gl1_b, l1_w, l1_b, p1, CCH, /*chan_major=*/0);
    proj_gemm_kernel<true><<<dim3(NEDGE / 256, CCH / 16), 256, 0, stream>>>(
        pl2, NEDGE, gl2_w, gl2_b, l2_w, l2_b, p2, CCH, /*chan_major=*/0);

    // 3) Dense sparse->dense scatter (duplicates sum)
    (void)hipMemsetAsync(P1, 0, (size_t)CCH * MTOT * sizeof(float), stream);
    (void)hipMemsetAsync(P2, 0, (size_t)CCH * MTOT * sizeof(float), stream);
    scatter_kernel<<<NEDGE, CCH, 0, stream>>>(edges1, p1, P1, NEDGE);
    scatter_kernel<<<NEDGE, CCH, 0, stream>>>(edges2, p2, P2, NEDGE);

    // 4) Per-channel 512^3 GEMMs: block1 = P1 @ ab1 ; block2 += ab2 @ P2^T
    chan_gemm_kernel<false, false><<<dim3(NN / 128, NN / 128, CCH), 256, 0, stream>>>(
        P1, ab1, Bsum);
    chan_gemm_kernel<true, true><<<dim3(NN / 128, NN / 128, CCH), 256, 0, stream>>>(
        ab2, P2, Bsum);

    // 5) LN over C + output projection, fused with g-gating
    final_kernel<<<MTOT / 16, 256, 0, stream>>>(
        Bsum, gbuf, lnc_w, lnc_b, las_w, las_b, out);
}